// TorchSMPL4Garment_22995254902807
// MI455X (gfx1250) — compile-verified
//
#include <hip/hip_runtime.h>
#include <hip/hip_bf16.h>

#define V_N   27554
#define V3    (3 * V_N)
#define NJ    24
#define NB    256
#define KP    220   // virtual K: 207 pose | 10 beta | 1 const | 2 zero

typedef __attribute__((ext_vector_type(2))) float v2f;
typedef __attribute__((ext_vector_type(8))) float v8f;

__constant__ int c_par[NJ] = {0, 0, 0, 0, 1, 2, 3, 4, 5, 6, 7, 8, 9, 9, 9,
                              12, 13, 14, 16, 17, 18, 19, 20, 21};

// ---------------------------------------------------------------------------
// Kernel 1: Rodrigues per joint + pack F = [(R-I) flat(207) | beta(10) | 1 | 0 0]
// grid = NB blocks, 32 threads
// ---------------------------------------------------------------------------
__global__ __launch_bounds__(32) void k_prep(const float* __restrict__ theta,
                                             const float* __restrict__ beta,
                                             float* __restrict__ F,
                                             float* __restrict__ Rs) {
  int n = blockIdx.x;
  int t = threadIdx.x;
  if (t < NJ) {
    const float* th = theta + n * 72 + t * 3;
    float tx = th[0], ty = th[1], tz = th[2];
    float ax = tx + 1e-8f, ay = ty + 1e-8f, az = tz + 1e-8f;
    float ang = sqrtf(ax * ax + ay * ay + az * az);
    float inv = 1.0f / ang;
    float nx = tx * inv, ny = ty * inv, nz = tz * inv;
    float h = 0.5f * ang;
    float qw = cosf(h), s = sinf(h);
    float qx = s * nx, qy = s * ny, qz = s * nz;
    float qn = rsqrtf(qw * qw + qx * qx + qy * qy + qz * qz);
    qw *= qn; qx *= qn; qy *= qn; qz *= qn;
    float R[9];
    R[0] = qw * qw + qx * qx - qy * qy - qz * qz;
    R[1] = 2.f * (qx * qy - qw * qz);
    R[2] = 2.f * (qw * qy + qx * qz);
    R[3] = 2.f * (qw * qz + qx * qy);
    R[4] = qw * qw - qx * qx + qy * qy - qz * qz;
    R[5] = 2.f * (qy * qz - qw * qx);
    R[6] = 2.f * (qx * qz - qw * qy);
    R[7] = 2.f * (qw * qx + qy * qz);
    R[8] = qw * qw - qx * qx - qy * qy + qz * qz;
#pragma unroll
    for (int e = 0; e < 9; ++e) Rs[n * (NJ * 9) + t * 9 + e] = R[e];
    if (t >= 1) {  // pose_feature occupies F cols 0..206
#pragma unroll
      for (int e = 0; e < 9; ++e) {
        float d = (e == 0 || e == 4 || e == 8) ? 1.0f : 0.0f;
        F[(size_t)n * KP + (t - 1) * 9 + e] = R[e] - d;
      }
    }
  }
  if (t < 10) F[(size_t)n * KP + 207 + t] = beta[n * 10 + t];  // cols 207..216
  if (t == 30) F[(size_t)n * KP + 217] = 1.0f;
  if (t == 31) {
    F[(size_t)n * KP + 218] = 0.0f;
    F[(size_t)n * KP + 219] = 0.0f;
  }
}

// ---------------------------------------------------------------------------
// Kernel 2: Jd[k][j][c] = sum_v shapedirs[k][3v+c] * Jreg[v][j]
//           Jt[j][c]    = sum_v v_template[3v+c]   * Jreg[v][j]
// ---------------------------------------------------------------------------
__global__ __launch_bounds__(256) void k_jdirs(const float* __restrict__ sh,
                                               const float* __restrict__ vt,
                                               const float* __restrict__ jr,
                                               float* __restrict__ Jd,
                                               float* __restrict__ Jt) {
  int b = blockIdx.x;
  int j = b / 3, c = b % 3;
  int tid = threadIdx.x;
  float acc[11];
#pragma unroll
  for (int a = 0; a < 11; ++a) acc[a] = 0.0f;
  for (int v = tid; v < V_N; v += 256) {
    float r = jr[v * NJ + j];
    int col = v * 3 + c;
#pragma unroll
    for (int k = 0; k < 10; ++k) acc[k] += sh[(size_t)k * V3 + col] * r;
    acc[10] += vt[col] * r;
  }
  __shared__ float red[256];
  for (int a = 0; a < 11; ++a) {
    red[tid] = acc[a];
    __syncthreads();
    for (int s = 128; s > 0; s >>= 1) {
      if (tid < s) red[tid] += red[tid + s];
      __syncthreads();
    }
    if (tid == 0) {
      if (a < 10) Jd[a * (NJ * 3) + j * 3 + c] = red[0];
      else        Jt[j * 3 + c] = red[0];
    }
    __syncthreads();
  }
}

// ---------------------------------------------------------------------------
// Kernel 3: J[n][j][c] = Jt[j][c] + sum_k beta[n][k] * Jd[k][j][c]
// ---------------------------------------------------------------------------
__global__ __launch_bounds__(256) void k_joints(const float* __restrict__ beta,
                                                const float* __restrict__ Jd,
                                                const float* __restrict__ Jt,
                                                float* __restrict__ Jb) {
  int idx = blockIdx.x * 256 + threadIdx.x;
  if (idx >= NB * NJ * 3) return;
  int n = idx / (NJ * 3);
  int rem = idx % (NJ * 3);
  float acc = Jt[rem];
#pragma unroll
  for (int k = 0; k < 10; ++k) acc += beta[n * 10 + k] * Jd[k * (NJ * 3) + rem];
  Jb[idx] = acc;
}

// ---------------------------------------------------------------------------
// Kernel 4: kinematic chain -> A transforms (top 3 rows, 12 floats/joint)
// ---------------------------------------------------------------------------
__global__ __launch_bounds__(256) void k_chain(const float* __restrict__ Rs,
                                               const float* __restrict__ Jb,
                                               float* __restrict__ Wb,
                                               float* __restrict__ Ab) {
  int n = blockIdx.x * blockDim.x + threadIdx.x;
  if (n >= NB) return;
  const float* Jn = Jb + n * (NJ * 3);
  for (int j = 0; j < NJ; ++j) {
    float R[9];
#pragma unroll
    for (int e = 0; e < 9; ++e) R[e] = Rs[n * (NJ * 9) + j * 9 + e];
    float Wc[12];
    if (j == 0) {
#pragma unroll
      for (int r = 0; r < 3; ++r) {
        Wc[r * 4 + 0] = R[r * 3 + 0];
        Wc[r * 4 + 1] = R[r * 3 + 1];
        Wc[r * 4 + 2] = R[r * 3 + 2];
        Wc[r * 4 + 3] = Jn[r];
      }
    } else {
      int p = c_par[j];
      float tx = Jn[j * 3 + 0] - Jn[p * 3 + 0];
      float ty = Jn[j * 3 + 1] - Jn[p * 3 + 1];
      float tz = Jn[j * 3 + 2] - Jn[p * 3 + 2];
      const float* Wp = Wb + (size_t)n * (NJ * 12) + p * 12;
#pragma unroll
      for (int r = 0; r < 3; ++r) {
        float a0 = Wp[r * 4 + 0], a1 = Wp[r * 4 + 1], a2 = Wp[r * 4 + 2];
#pragma unroll
        for (int cc = 0; cc < 3; ++cc)
          Wc[r * 4 + cc] = a0 * R[0 * 3 + cc] + a1 * R[1 * 3 + cc] + a2 * R[2 * 3 + cc];
        Wc[r * 4 + 3] = a0 * tx + a1 * ty + a2 * tz + Wp[r * 4 + 3];
      }
    }
#pragma unroll
    for (int e = 0; e < 12; ++e) Wb[(size_t)n * (NJ * 12) + j * 12 + e] = Wc[e];
    float jx = Jn[j * 3 + 0], jy = Jn[j * 3 + 1], jz = Jn[j * 3 + 2];
#pragma unroll
    for (int r = 0; r < 3; ++r) {
      Ab[(size_t)n * (NJ * 12) + j * 12 + r * 4 + 0] = Wc[r * 4 + 0];
      Ab[(size_t)n * (NJ * 12) + j * 12 + r * 4 + 1] = Wc[r * 4 + 1];
      Ab[(size_t)n * (NJ * 12) + j * 12 + r * 4 + 2] = Wc[r * 4 + 2];
      Ab[(size_t)n * (NJ * 12) + j * 12 + r * 4 + 3] =
          Wc[r * 4 + 3] - (Wc[r * 4 + 0] * jx + Wc[r * 4 + 1] * jy + Wc[r * 4 + 2] * jz);
    }
  }
}

#define WMMA3(a, bx, by, bz)                                                   \
  accX = __builtin_amdgcn_wmma_f32_16x16x4_f32(false, a, false, bx, (short)0,  \
                                               accX, false, false);            \
  accY = __builtin_amdgcn_wmma_f32_16x16x4_f32(false, a, false, by, (short)0,  \
                                               accY, false, false);            \
  accZ = __builtin_amdgcn_wmma_f32_16x16x4_f32(false, a, false, bz, (short)0,  \
                                               accZ, false, false);

// ---------------------------------------------------------------------------
// Kernel 5 (fused main): 16-batch x 16-vertex tile per wave.
//   v_posed(x,y,z) via v_wmma_f32_16x16x4_f32 over virtual K=220
//   (K order: posedirs 0..206 | shapedirs 0..9 | v_template | 0 | 0),
//   then per-lane skinning against LDS-staged A tile.
// block = 128 (4 waves), grid = (ceil(V/64), NB/16)
// ---------------------------------------------------------------------------
__global__ __launch_bounds__(128) void k_main(const float* __restrict__ F,
                                              const float* __restrict__ sh,
                                              const float* __restrict__ po,
                                              const float* __restrict__ vt,
                                              const float* __restrict__ wts,
                                              const float* __restrict__ Ab,
                                              float* __restrict__ out) {
  __shared__ float sA[16 * NJ * 12];  // 18 KB

  int m0 = blockIdx.y * 16;
  int wave = threadIdx.x >> 5;
  int lane = threadIdx.x & 31;
  int v0 = blockIdx.x * 64 + wave * 16;

  for (int i = threadIdx.x; i < 16 * NJ * 12; i += 128)
    sA[i] = Ab[(size_t)(m0 + i / (NJ * 12)) * (NJ * 12) + (i % (NJ * 12))];
  __syncthreads();

  int half = lane >> 4;  // 0: K pair {k,k+1}, M rows r; 1: K pair {k+2,k+3}, rows r+8
  int l16 = lane & 15;
  int vcol = v0 + l16;
  int vc = vcol < V_N ? vcol : V_N - 1;  // clamp for safe OOB loads
  int colb = vc * 3;

  const float* Frow = F + (size_t)(m0 + l16) * KP;
  int koff = half * 2;

  v8f accX = {}, accY = {}, accZ = {};

  // ---- segment 1: k = 0..203, pure posedirs rows (branch-free) ----
#pragma unroll 2
  for (int k = 0; k < 204; k += 4) {
    int kb = k + koff;
    float2 av = *(const float2*)(Frow + kb);
    v2f a; a[0] = av.x; a[1] = av.y;
    const float* r0 = po + (size_t)kb * V3 + colb;
    const float* r1 = r0 + V3;
    v2f bx, by, bz;
    bx[0] = r0[0]; by[0] = r0[1]; bz[0] = r0[2];
    bx[1] = r1[0]; by[1] = r1[1]; bz[1] = r1[2];
    WMMA3(a, bx, by, bz)
  }

  // ---- segment 2: k = 204, rows 204..207 = {po 204, po 205, po 206, sh 0}
  {
    int kb = 204 + koff;
    float2 av = *(const float2*)(Frow + kb);
    v2f a; a[0] = av.x; a[1] = av.y;
    const float* r0 = po + (size_t)kb * V3 + colb;                    // 204 / 206
    const float* r1 = half ? (sh + colb) : (po + (size_t)205 * V3 + colb);
    v2f bx, by, bz;
    bx[0] = r0[0]; by[0] = r0[1]; bz[0] = r0[2];
    bx[1] = r1[0]; by[1] = r1[1]; bz[1] = r1[2];
    WMMA3(a, bx, by, bz)
  }

  // ---- segment 3: k = 208,212, rows 208..215 = shapedirs rows 1..8 ----
#pragma unroll
  for (int k = 208; k <= 212; k += 4) {
    int kb = k + koff;
    float2 av = *(const float2*)(Frow + kb);
    v2f a; a[0] = av.x; a[1] = av.y;
    const float* r0 = sh + (size_t)(kb - 207) * V3 + colb;
    const float* r1 = r0 + V3;
    v2f bx, by, bz;
    bx[0] = r0[0]; by[0] = r0[1]; bz[0] = r0[2];
    bx[1] = r1[0]; by[1] = r1[1]; bz[1] = r1[2];
    WMMA3(a, bx, by, bz)
  }

  // ---- segment 4: k = 216, rows 216..219 = {sh 9, v_template, 0, 0} ----
  {
    float2 av = *(const float2*)(Frow + 216 + koff);
    v2f a; a[0] = av.x; a[1] = av.y;
    const float* r9 = sh + (size_t)9 * V3 + colb;
    float s9x = r9[0], s9y = r9[1], s9z = r9[2];
    float tvx = vt[colb], tvy = vt[colb + 1], tvz = vt[colb + 2];
    v2f bx, by, bz;
    bx[0] = half ? 0.0f : s9x; by[0] = half ? 0.0f : s9y; bz[0] = half ? 0.0f : s9z;
    bx[1] = half ? 0.0f : tvx; by[1] = half ? 0.0f : tvy; bz[1] = half ? 0.0f : tvz;
    WMMA3(a, bx, by, bz)
  }

  // ---- per-lane skinning: column vcol, rows (r + 8*half) ----
  float w[NJ];
  const float4* wrow = (const float4*)(wts + (size_t)vc * NJ);
#pragma unroll
  for (int q = 0; q < 6; ++q) {
    float4 t = wrow[q];
    w[q * 4 + 0] = t.x; w[q * 4 + 1] = t.y; w[q * 4 + 2] = t.z; w[q * 4 + 3] = t.w;
  }

#pragma unroll
  for (int r = 0; r < 8; ++r) {
    int nl = r + 8 * half;
    float x = accX[r], y = accY[r], z = accZ[r];
    float T[12];
#pragma unroll
    for (int e = 0; e < 12; ++e) T[e] = 0.0f;
    const float* Ar = sA + nl * (NJ * 12);
    for (int j = 0; j < NJ; ++j) {
      float wj = w[j];
      const float4* Aj = (const float4*)(Ar + j * 12);
      float4 a0 = Aj[0], a1 = Aj[1], a2 = Aj[2];
      T[0] = fmaf(wj, a0.x, T[0]);  T[1] = fmaf(wj, a0.y, T[1]);
      T[2] = fmaf(wj, a0.z, T[2]);  T[3] = fmaf(wj, a0.w, T[3]);
      T[4] = fmaf(wj, a1.x, T[4]);  T[5] = fmaf(wj, a1.y, T[5]);
      T[6] = fmaf(wj, a1.z, T[6]);  T[7] = fmaf(wj, a1.w, T[7]);
      T[8] = fmaf(wj, a2.x, T[8]);  T[9] = fmaf(wj, a2.y, T[9]);
      T[10] = fmaf(wj, a2.z, T[10]); T[11] = fmaf(wj, a2.w, T[11]);
    }
    if (vcol < V_N) {
      size_t o = (size_t)(m0 + nl) * V3 + (size_t)vcol * 3;
      out[o + 0] = T[0] * x + T[1] * y + T[2] * z + T[3];
      out[o + 1] = T[4] * x + T[5] * y + T[6] * z + T[7];
      out[o + 2] = T[8] * x + T[9] * y + T[10] * z + T[11];
    }
  }
}

// ---------------------------------------------------------------------------
extern "C" void kernel_launch(void* const* d_in, const int* in_sizes, int n_in,
                              void* d_out, int out_size, void* d_ws, size_t ws_size,
                              hipStream_t stream) {
  const float* theta = (const float*)d_in[0];   // (256,72)
  const float* beta  = (const float*)d_in[1];   // (256,10)
  const float* vtem  = (const float*)d_in[2];   // (V,3)
  const float* shd   = (const float*)d_in[3];   // (10, V*3)
  const float* jreg  = (const float*)d_in[4];   // (V, 24)
  const float* pod   = (const float*)d_in[5];   // (207, V*3)
  const float* wts   = (const float*)d_in[6];   // (V, 24)
  float* out = (float*)d_out;

  float* F  = (float*)d_ws;              // NB*KP
  float* Rs = F + (size_t)NB * KP;       // NB*NJ*9
  float* Jd = Rs + (size_t)NB * NJ * 9;  // 10*NJ*3
  float* Jt = Jd + 10 * NJ * 3;          // NJ*3
  float* Jb = Jt + NJ * 3;               // NB*NJ*3
  float* Wb = Jb + (size_t)NB * NJ * 3;  // NB*NJ*12
  float* Ab = Wb + (size_t)NB * NJ * 12; // NB*NJ*12

  k_prep<<<NB, 32, 0, stream>>>(theta, beta, F, Rs);
  k_jdirs<<<NJ * 3, 256, 0, stream>>>(shd, vtem, jreg, Jd, Jt);
  k_joints<<<(NB * NJ * 3 + 255) / 256, 256, 0, stream>>>(beta, Jd, Jt, Jb);
  k_chain<<<1, NB, 0, stream>>>(Rs, Jb, Wb, Ab);
  k_main<<<dim3((V_N + 63) / 64, NB / 16), 128, 0, stream>>>(F, shd, pod, vtem,
                                                             wts, Ab, out);
}